// SWD5_66932770341577
// MI455X (gfx1250) — compile-verified
//
#include <hip/hip_runtime.h>
#include <hip/hip_bf16.h>
#include <cstdint>

typedef __attribute__((ext_vector_type(2))) float v2f;
typedef __attribute__((ext_vector_type(4))) float v4f;
typedef __attribute__((ext_vector_type(8))) float v8f;

#define Bb 2
#define Hh 8
#define Ll 2048
#define Dd 64
#define BH (Bb * Hh)

// ---------------------------------------------------------------------------
// Zero the attn_weights region of d_out (scatter target).
// ---------------------------------------------------------------------------
__global__ __launch_bounds__(256) void SWD5_zero_kernel(float* __restrict__ p, size_t n) {
    size_t i = (size_t)blockIdx.x * blockDim.x + threadIdx.x;
    const size_t stride = (size_t)gridDim.x * blockDim.x;
    const size_t n4 = n >> 2;
    v4f* p4 = (v4f*)p;
    for (; i < n4; i += stride) p4[i] = (v4f){0.0f, 0.0f, 0.0f, 0.0f};
}

// ---------------------------------------------------------------------------
// Per-(b,h,d) bitonic argsort of q[:,d] and k[:,d] (L=2048) in LDS, then
// scatter-add exp(-(qs-ks)^2)/D into attn[bh, q_idx[i], k_idx[i]].
// Keys: sortable-float-bits in high 32, index in low 32 (stable ascending).
// ---------------------------------------------------------------------------
__device__ __forceinline__ uint32_t f32_to_sortable(float f) {
    uint32_t u = __float_as_uint(f);
    return (u & 0x80000000u) ? ~u : (u | 0x80000000u);
}
__device__ __forceinline__ float sortable_to_f32(uint32_t s) {
    uint32_t u = (s & 0x80000000u) ? (s & 0x7FFFFFFFu) : ~s;
    return __uint_as_float(u);
}

__global__ __launch_bounds__(256) void SWD5_sort_scatter_kernel(
        const float* __restrict__ q, const float* __restrict__ k,
        float* __restrict__ attn) {
    __shared__ uint64_t sq[Ll];
    __shared__ uint64_t sk[Ll];
    const int tid = threadIdx.x;
    const int bh  = blockIdx.x / Dd;
    const int d   = blockIdx.x % Dd;
    const float* qp = q + (size_t)bh * Ll * Dd + d;
    const float* kp = k + (size_t)bh * Ll * Dd + d;

    for (int i = tid; i < Ll; i += 256) {
        sq[i] = ((uint64_t)f32_to_sortable(qp[(size_t)i * Dd]) << 32) | (uint32_t)i;
        sk[i] = ((uint64_t)f32_to_sortable(kp[(size_t)i * Dd]) << 32) | (uint32_t)i;
    }
    __syncthreads();

    // Bitonic sort (ascending), both arrays share the control flow.
    for (int size = 2; size <= Ll; size <<= 1) {
        for (int j = size >> 1; j > 0; j >>= 1) {
            for (int i = tid; i < Ll; i += 256) {
                const int ixj = i ^ j;
                if (ixj > i) {
                    const bool up = ((i & size) == 0);
                    uint64_t a = sq[i], b = sq[ixj];
                    if ((a > b) == up) { sq[i] = b; sq[ixj] = a; }
                    uint64_t c = sk[i], e = sk[ixj];
                    if ((c > e) == up) { sk[i] = e; sk[ixj] = c; }
                }
            }
            __syncthreads();
        }
    }

    float* ap = attn + (size_t)bh * Ll * Ll;
    for (int i = tid; i < Ll; i += 256) {
        const uint64_t eq = sq[i], ek = sk[i];
        const float qv = sortable_to_f32((uint32_t)(eq >> 32));
        const float kv = sortable_to_f32((uint32_t)(ek >> 32));
        const int qi = (int)(eq & 0xFFFFFFFFu);
        const int ki = (int)(ek & 0xFFFFFFFFu);
        const float diff = qv - kv;
        const float w = __expf(-diff * diff) * (1.0f / (float)Dd);
        atomicAdd(ap + (size_t)qi * Ll + ki, w);  // global_atomic_add_f32
    }
}

// ---------------------------------------------------------------------------
// attn = attn + attn^T, in place; each (i,j) pair with j>=i handled once.
// ---------------------------------------------------------------------------
__global__ __launch_bounds__(256) void SWD5_sym_kernel(float* __restrict__ attn) {
    const int j  = blockIdx.x * 16 + (threadIdx.x & 15);
    const int i  = blockIdx.y * 16 + (threadIdx.x >> 4);
    const int bh = blockIdx.z;
    if (j < i) return;
    float* ap = attn + (size_t)bh * Ll * Ll;
    const float a = ap[(size_t)i * Ll + j];
    const float b = ap[(size_t)j * Ll + i];
    const float s = a + b;
    ap[(size_t)i * Ll + j] = s;
    ap[(size_t)j * Ll + i] = s;
}

// ---------------------------------------------------------------------------
// out[bh] = where(mask,0,attn[bh]) @ v[bh] via V_WMMA_F32_16X16X4_F32 with
// LDS tiling. Block = 256 threads = 8 wave32s covering 32 rows x 64 cols
// (2 M-tiles x 4 N-tiles). K chunked by 64:
//   As: 32x64 masked attn, row-major, stride 68 floats (bank-conflict-free
//       b64 A-fragment reads: row r -> banks 4r,4r+1 / 4r+2,4r+3).
//   Bs: 64x64 of v, stored transposed [col][k] with the same padded stride
//       so B fragments are contiguous b64 LDS reads, same bank pattern.
// A frag (16x4 f32): lanes 0-15 hold K=0,1 in the 2 VGPRs, lanes 16-31 K=2,3.
// C/D (16x16 f32): VGPR r -> M = r (+8 for lanes 16-31), N = lane&15.
// ---------------------------------------------------------------------------
__global__ __launch_bounds__(256) void SWD5_gemm_kernel(
        const float* __restrict__ attn, const unsigned char* __restrict__ mask,
        const float* __restrict__ v, float* __restrict__ out) {
    constexpr int KC  = 64;   // K chunk
    constexpr int LDA = 68;   // padded LDS stride in floats
    __shared__ float As[32 * LDA];
    __shared__ float Bs[64 * LDA];

    const int tid     = threadIdx.x;
    const int bh      = blockIdx.x >> 6;        // 64 strips of 32 rows
    const int rowBase = (blockIdx.x & 63) * 32;
    const int wave    = tid >> 5;
    const int lane    = tid & 31;
    const int mtile   = wave >> 2;              // 0..1
    const int ntile   = wave & 3;               // 0..3
    const int m       = lane & 15;
    const int khalf   = (lane >> 4) << 1;       // 0 or 2

    // Staging coordinates: A -> 8 floats/thread, B -> 16 floats/thread.
    const int ar  = tid >> 3;                   // 0..31
    const int ac8 = (tid & 7) * 8;              // 0..56
    const float*         Ag = attn + ((size_t)bh * Ll + rowBase + ar) * Ll;
    const unsigned char* Mg = mask + (size_t)(rowBase + ar) * Ll;
    const float*         Vb = v + (size_t)bh * Ll * Dd;

    const float* Asm = &As[(mtile * 16 + m) * LDA];
    const float* Bsm = &Bs[(ntile * 16 + m) * LDA];

    v8f acc = {};
    for (int k0 = 0; k0 < Ll; k0 += KC) {
        // ---- stage masked A tile (mask applied once per block) ----
        const float4  a0 = *(const float4*)(Ag + k0 + ac8);
        const float4  a1 = *(const float4*)(Ag + k0 + ac8 + 4);
        const uint64_t mb = *(const uint64_t*)(Mg + k0 + ac8);
        float va[8] = {a0.x, a0.y, a0.z, a0.w, a1.x, a1.y, a1.z, a1.w};
#pragma unroll
        for (int e = 0; e < 8; ++e)
            if ((mb >> (8 * e)) & 0xFFull) va[e] = 0.0f;
        *(v4f*)&As[ar * LDA + ac8]     = (v4f){va[0], va[1], va[2], va[3]};
        *(v4f*)&As[ar * LDA + ac8 + 4] = (v4f){va[4], va[5], va[6], va[7]};

        // ---- stage B tile of v, transposed to [col][k] ----
#pragma unroll
        for (int it = 0; it < 4; ++it) {
            const int li = tid + it * 256;
            const int rr = li >> 4;             // k within chunk: 0..63
            const int cc = (li & 15) * 4;       // col: 0..60
            const float4 bv = *(const float4*)(Vb + (size_t)(k0 + rr) * Dd + cc);
            Bs[(cc + 0) * LDA + rr] = bv.x;
            Bs[(cc + 1) * LDA + rr] = bv.y;
            Bs[(cc + 2) * LDA + rr] = bv.z;
            Bs[(cc + 3) * LDA + rr] = bv.w;
        }
        __syncthreads();

        // ---- 16 WMMAs over the chunk, fragments straight from LDS ----
#pragma unroll
        for (int kk4 = 0; kk4 < KC; kk4 += 4) {
            const int kk = kk4 + khalf;
            const v2f Af = *(const v2f*)(Asm + kk);
            const v2f Bf = *(const v2f*)(Bsm + kk);
            acc = __builtin_amdgcn_wmma_f32_16x16x4_f32(
                      false, Af, false, Bf, (short)0, acc, false, false);
        }
        __syncthreads();
    }

    const int mOut = rowBase + mtile * 16 + ((lane >> 4) << 3);
    float* Obh = out + (size_t)bh * Ll * Dd;
    const int col = ntile * 16 + m;
#pragma unroll
    for (int r = 0; r < 8; ++r)
        Obh[(size_t)(mOut + r) * Dd + col] = acc[r];
}

// ---------------------------------------------------------------------------
extern "C" void kernel_launch(void* const* d_in, const int* in_sizes, int n_in,
                              void* d_out, int out_size, void* d_ws, size_t ws_size,
                              hipStream_t stream) {
    const float* q = (const float*)d_in[0];
    const float* k = (const float*)d_in[1];
    const float* v = (const float*)d_in[2];
    const unsigned char* mask = (const unsigned char*)d_in[3];  // bool, 1B/elem

    float* out  = (float*)d_out;                       // (B,H,L,D)
    float* attn = out + (size_t)BH * Ll * Dd;          // (B,H,L,L)

    SWD5_zero_kernel<<<4096, 256, 0, stream>>>(attn, (size_t)BH * Ll * Ll);
    SWD5_sort_scatter_kernel<<<BH * Dd, 256, 0, stream>>>(q, k, attn);
    dim3 symGrid(Ll / 16, Ll / 16, BH);
    SWD5_sym_kernel<<<symGrid, 256, 0, stream>>>(attn);
    SWD5_gemm_kernel<<<BH * (Ll / 32), 256, 0, stream>>>(attn, mask, v, out);
}